// WindAdjacency_43894565765671
// MI455X (gfx1250) — compile-verified
//
#include <hip/hip_runtime.h>
#include <cmath>
#include <cstdint>

// ---------------------------------------------------------------------------
// WindAdjacency on MI455X (gfx1250)
//   out[b,i,j] = mask * speed[b,i] * exp(max(cos(ang),0) - D[i,j]/R) / rowsum
//   ang = wrap(Theta[i,j] - wrap(dir[b,i]+pi) + pi) - pi,  mask = |ang|<=pi/4
// Memory-bound: ~134 MB out + ~32 MB in => ~7us @ 23.3 TB/s.
// One workgroup per row i, loops over all B batches so D/Theta rows are read
// from HBM once (staged via CDNA5 async global->LDS, then register-cached).
// ---------------------------------------------------------------------------

#define BLOCK 256

__device__ __forceinline__ float pmod_2pi(float x) {
    // Python-% semantics: result in [0, 2pi)
    const float TWO_PI     = 6.28318530717958647692f;
    const float INV_TWO_PI = 0.15915494309189533577f;
    return x - floorf(x * INV_TWO_PI) * TWO_PI;
}

__device__ __forceinline__ float wind_elem(float th, float d, float wto,
                                           float ws, bool diag) {
    const float PI_F  = 3.14159265358979323846f;
    const float CONE  = 0.78539816339744830962f;  // pi/4
    const float INV_R = 1.0f / 150.0f;
    float ang   = pmod_2pi(th - wto + PI_F) - PI_F;     // [-pi, pi)
    float c     = __cosf(ang);                          // v_cos_f32
    float align = fmaxf(c, 0.0f);
    float v     = ws * __expf(align - d * INV_R);       // v_exp_f32
    v = (fabsf(ang) <= CONE) ? v : 0.0f;
    return diag ? 0.0f : v;
}

#if defined(__gfx1250__) && __has_builtin(__builtin_amdgcn_global_load_async_to_lds_b128)
#define USE_ASYNC_LDS 1

typedef __attribute__((ext_vector_type(4))) int v4i;
typedef __attribute__((address_space(1))) v4i* gv4i_p;  // global (AS1) int4*
typedef __attribute__((address_space(3))) v4i* lv4i_p;  // LDS    (AS3) int4*

// Generic->AS1: identity value. Reinterpret via integer (always legal).
__device__ __forceinline__ gv4i_p as_global_v4i(const void* p) {
    return (gv4i_p)(unsigned long long)(uintptr_t)p;
}
// Generic->AS3: LDS byte offset lives in the low 32 bits of the flat address
// AS3 pointers are 32-bit.
__device__ __forceinline__ lv4i_p as_lds_v4i(const void* p) {
    return (lv4i_p)(unsigned int)(uintptr_t)p;
}
#endif

template <int CHUNKS>
__global__ __launch_bounds__(BLOCK)
void wind_adj_kernel(const float* __restrict__ wind,  // [B,N,2]
                     const float* __restrict__ Dm,    // [N,N]
                     const float* __restrict__ Th,    // [N,N]
                     float* __restrict__ out,         // [B,N,N]
                     int B, int N) {
    __shared__ __align__(16) float ldsD[CHUNKS * BLOCK * 4];
    __shared__ __align__(16) float ldsT[CHUNKS * BLOCK * 4];
    __shared__ float redbuf[BLOCK / 32];

    const int tid = threadIdx.x;
    const int i   = blockIdx.x;
    const long long rowOff = (long long)i * N;

    // ---- stage row i of D and Theta into LDS (CDNA5 async DMA path) ----
#ifdef USE_ASYNC_LDS
#pragma unroll
    for (int c = 0; c < CHUNKS; ++c) {
        const int e = (c * BLOCK + tid) * 4;
        if (e < N) {
            __builtin_amdgcn_global_load_async_to_lds_b128(
                as_global_v4i(Dm + rowOff + e), as_lds_v4i(&ldsD[e]), 0, 0);
            __builtin_amdgcn_global_load_async_to_lds_b128(
                as_global_v4i(Th + rowOff + e), as_lds_v4i(&ldsT[e]), 0, 0);
        }
    }
#if __has_builtin(__builtin_amdgcn_s_wait_asynccnt)
    __builtin_amdgcn_s_wait_asynccnt(0);
#else
    asm volatile("s_wait_asynccnt 0" ::: "memory");
#endif
#else
    // Fallback: synchronous copy through registers (still coalesced B128).
#pragma unroll
    for (int c = 0; c < CHUNKS; ++c) {
        const int e = (c * BLOCK + tid) * 4;
        if (e < N) {
            *(float4*)&ldsD[e] = *(const float4*)(Dm + rowOff + e);
            *(float4*)&ldsT[e] = *(const float4*)(Th + rowOff + e);
        }
    }
#endif

    // Each wave consumed only data it loaded itself -> no barrier needed.
    // Pull the row into registers; reused across all B batches.
    float4 dv[CHUNKS], tv[CHUNKS];
#pragma unroll
    for (int c = 0; c < CHUNKS; ++c) {
        const int e = (c * BLOCK + tid) * 4;
        if (e < N) {
            dv[c] = *(const float4*)&ldsD[e];
            tv[c] = *(const float4*)&ldsT[e];
        } else {
            dv[c] = make_float4(0.f, 0.f, 0.f, 0.f);
            tv[c] = make_float4(0.f, 0.f, 0.f, 0.f);
        }
    }

    for (int b = 0; b < B; ++b) {
        // Per-(b,i) scalars: all lanes hit the same cacheline.
        const float2 wf = *(const float2*)(wind + ((long long)b * N + i) * 2);
        const float ws  = wf.x;
        const float wto = pmod_2pi(wf.y + 3.14159265358979323846f);

        // ---- compute base values, keep in registers, accumulate row sum ----
        float4 bv[CHUNKS];
        float psum = 0.0f;
#pragma unroll
        for (int c = 0; c < CHUNKS; ++c) {
            const int e = (c * BLOCK + tid) * 4;
            float4 r;
            r.x = wind_elem(tv[c].x, dv[c].x, wto, ws, (e + 0) == i);
            r.y = wind_elem(tv[c].y, dv[c].y, wto, ws, (e + 1) == i);
            r.z = wind_elem(tv[c].z, dv[c].z, wto, ws, (e + 2) == i);
            r.w = wind_elem(tv[c].w, dv[c].w, wto, ws, (e + 3) == i);
            bv[c] = r;
            if (e < N) psum += (r.x + r.y) + (r.z + r.w);
        }

        // ---- wave32 butterfly reduction, then cross-wave via LDS ----
        float s = psum;
#pragma unroll
        for (int off = 16; off > 0; off >>= 1) s += __shfl_xor(s, off, 32);
        if ((tid & 31) == 0) redbuf[tid >> 5] = s;
        __syncthreads();
        float tot = 0.0f;
#pragma unroll
        for (int w = 0; w < BLOCK / 32; ++w) tot += redbuf[w];
        const float inv = 1.0f / (tot + 1e-8f);

        // ---- single coalesced write pass ----
        float* orow = out + ((long long)b * N + i) * (long long)N;
#pragma unroll
        for (int c = 0; c < CHUNKS; ++c) {
            const int e = (c * BLOCK + tid) * 4;
            if (e < N) {
                float4 o;
                o.x = bv[c].x * inv;
                o.y = bv[c].y * inv;
                o.z = bv[c].z * inv;
                o.w = bv[c].w * inv;
                *(float4*)(orow + e) = o;
            }
        }
        __syncthreads();  // protect redbuf before next batch reuses it
    }
}

extern "C" void kernel_launch(void* const* d_in, const int* in_sizes, int n_in,
                              void* d_out, int out_size, void* d_ws, size_t ws_size,
                              hipStream_t stream) {
    const float* wind = (const float*)d_in[0];   // [B,N,2] f32
    const float* Dm   = (const float*)d_in[1];   // [N,N]   f32
    const float* Th   = (const float*)d_in[2];   // [N,N]   f32
    float* out        = (float*)d_out;           // [B,N,N] f32

    const int N = (int)(std::sqrt((double)in_sizes[1]) + 0.5);
    const int B = in_sizes[0] / (2 * N);

    const int chunks = (N + BLOCK * 4 - 1) / (BLOCK * 4);
    dim3 grid(N), block(BLOCK);
    if (chunks <= 1) {
        wind_adj_kernel<1><<<grid, block, 0, stream>>>(wind, Dm, Th, out, B, N);
    } else if (chunks <= 2) {   // N = 2048 (reference shape)
        wind_adj_kernel<2><<<grid, block, 0, stream>>>(wind, Dm, Th, out, B, N);
    } else if (chunks <= 4) {
        wind_adj_kernel<4><<<grid, block, 0, stream>>>(wind, Dm, Th, out, B, N);
    } else {
        wind_adj_kernel<8><<<grid, block, 0, stream>>>(wind, Dm, Th, out, B, N);
    }
}